// LDM_34875134443547
// MI455X (gfx1250) — compile-verified
//
#include <hip/hip_runtime.h>
#include <hip/hip_bf16.h>
#include <math.h>

// CDNA5 / gfx1250 Kalman filter (LDM.compute_forwards).
// One wave32 per sequence; all dense algebra on V_WMMA_F32_16X16X4_F32.

typedef __attribute__((ext_vector_type(2))) float v2f;
typedef __attribute__((ext_vector_type(8))) float v8f;

#define DIM_X 16
#define DIM_A 32
#define NUM_SEQ 256
#define NUM_STEPS 1000

// ---- WMMA fragment helpers (f32, 16x16 tiles, K processed 4 at a time) ----
// A-matrix 16x4 chunk layout (ISA 7.12.2): lanes 0-15 hold K=4kc+{0,1},
// lanes 16-31 hold K=4kc+{2,3}, M = lane&15.
__device__ __forceinline__ v2f ld_a(const float* M, int ld, int kc, int lane) {
  int m = lane & 15;
  int k0 = 4 * kc + ((lane & 16) ? 2 : 0);
  v2f a; a.x = M[m * ld + k0]; a.y = M[m * ld + k0 + 1]; return a;
}
// A-fragment of M^T (A[m][k] = M[k][m])
__device__ __forceinline__ v2f ld_aT(const float* M, int ld, int kc, int lane) {
  int m = lane & 15;
  int k0 = 4 * kc + ((lane & 16) ? 2 : 0);
  v2f a; a.x = M[k0 * ld + m]; a.y = M[(k0 + 1) * ld + m]; return a;
}
// B-matrix 4x16 chunk: VGPR v holds row K=4kc+v (lanes 0-15) / K=4kc+v+2 (16-31).
__device__ __forceinline__ v2f ld_b(const float* M, int ld, int kc, int lane) {
  int n = lane & 15;
  int k0 = 4 * kc + ((lane & 16) ? 2 : 0);
  v2f b; b.x = M[k0 * ld + n]; b.y = M[(k0 + 1) * ld + n]; return b;
}
// B-fragment of M^T (B[k][n] = M[n][k])
__device__ __forceinline__ v2f ld_bT(const float* M, int ld, int kc, int lane) {
  int n = lane & 15;
  int k0 = 4 * kc + ((lane & 16) ? 2 : 0);
  v2f b; b.x = M[n * ld + k0]; b.y = M[n * ld + k0 + 1]; return b;
}
// C/D 16x16 f32 fragment: VGPR v = row v (lanes 0-15) / row v+8 (lanes 16-31).
__device__ __forceinline__ v8f ld_cd(const float* M, int ld, int lane) {
  int n = lane & 15; int r0 = (lane & 16) ? 8 : 0;
  v8f c;
#pragma unroll
  for (int v = 0; v < 8; ++v) c[v] = M[(r0 + v) * ld + n];
  return c;
}
__device__ __forceinline__ void st_cd(float* M, int ld, int lane, v8f c) {
  int n = lane & 15; int r0 = (lane & 16) ? 8 : 0;
#pragma unroll
  for (int v = 0; v < 8; ++v) M[(r0 + v) * ld + n] = c[v];
}
__device__ __forceinline__ v8f zero8() {
  v8f z;
#pragma unroll
  for (int v = 0; v < 8; ++v) z[v] = 0.0f;
  return z;
}
#define WMMA4(a, b, c) \
  __builtin_amdgcn_wmma_f32_16x16x4_f32(false, (a), false, (b), (short)0, (c), false, false)

__global__ __launch_bounds__(32) void ldm_kalman_kernel(
    const float* __restrict__ ga, const float* __restrict__ gmask,
    const float* __restrict__ gA, const float* __restrict__ gC,
    const float* __restrict__ gmu0, const float* __restrict__ gLam0,
    const float* __restrict__ gWlog, const float* __restrict__ gRlog,
    float* __restrict__ out) {
  const int lane = threadIdx.x;
  const int n = blockIdx.x;

  __shared__ float sP[16 * 17];    // Lambda_pred (state)
  __shared__ float sS[32 * 33];    // innovation covariance (augmented solve)
  __shared__ float sX[32 * 17];    // X = K^T (solve workspace)
  __shared__ float sCP[32 * 17];   // C @ P
  __shared__ float sT[16 * 17];    // Lambda_t
  __shared__ float sT2[16 * 17];   // A @ Lambda_t
  __shared__ float sMu[16];        // mu_pred (state)
  __shared__ float sMuT[16];       // mu_t
  __shared__ float sR[32];         // residual

  float* o_mu_pred = out;
  float* o_mu_t    = out + (size_t)NUM_STEPS * NUM_SEQ * DIM_X;
  float* o_Lp      = out + (size_t)2 * NUM_STEPS * NUM_SEQ * DIM_X;
  float* o_Lt      = o_Lp + (size_t)NUM_STEPS * NUM_SEQ * DIM_X * DIM_X;

  // ---- init state ----
#pragma unroll
  for (int v = 0; v < 8; ++v) {
    int idx = lane * 8 + v;                       // 256 elements of Lambda_0
    sP[(idx >> 4) * 17 + (idx & 15)] = gLam0[idx];
  }
  if (lane < 16) sMu[lane] = gmu0[lane];

  // ---- loop-invariant operands, staged once into register fragments ----
  float Crow[16];                                  // lane i owns C[i][:]
#pragma unroll
  for (int j = 0; j < 16; ++j) Crow[j] = gC[lane * 16 + j];
  float Arow[16];                                  // lanes<16: A[lane][:]
#pragma unroll
  for (int j = 0; j < 16; ++j) Arow[j] = gA[(lane & 15) * 16 + j];
  const float rdiag = expf(gRlog[lane]);           // diag(R)[lane]

  v2f aA[4], bAT[4], aC0[4], aC1[4], bCT0[4], bCT1[4];
#pragma unroll
  for (int kc = 0; kc < 4; ++kc) {
    aA[kc]   = ld_a (gA, 16, kc, lane);            // A as A-fragments
    bAT[kc]  = ld_bT(gA, 16, kc, lane);            // A^T as B-fragments
    aC0[kc]  = ld_a (gC,            16, kc, lane); // C rows 0-15
    aC1[kc]  = ld_a (gC + 16 * 16,  16, kc, lane); // C rows 16-31
    bCT0[kc] = ld_bT(gC,            16, kc, lane); // C^T cols 0-15
    bCT1[kc] = ld_bT(gC + 16 * 16,  16, kc, lane); // C^T cols 16-31
  }
  v8f wfrag;                                       // diag(exp(W_log)) as C-frag
  {
    int nc = lane & 15; int r0 = (lane & 16) ? 8 : 0;
    float wv = expf(gWlog[nc]);
#pragma unroll
    for (int v = 0; v < 8; ++v) wfrag[v] = ((r0 + v) == nc) ? wv : 0.0f;
  }
  __syncthreads();

  for (int t = 0; t < NUM_STEPS; ++t) {
    // ---- observation, residual r = a*m - C mu_pred (lane = a-dim) ----
    float a_l = ga[((size_t)n * NUM_STEPS + t) * DIM_A + lane];
    float m   = gmask[(size_t)n * NUM_STEPS + t];
    if (t + 1 < NUM_STEPS)
      __builtin_prefetch(&ga[((size_t)n * NUM_STEPS + t + 1) * DIM_A + lane], 0, 1);
    float ah = 0.0f;
#pragma unroll
    for (int j = 0; j < 16; ++j) ah += Crow[j] * sMu[j];
    sR[lane] = a_l * m - ah;

    // ---- CP = C @ P  (32x16, two WMMA tiles, K=16) ----
    {
      v2f bP[4];
#pragma unroll
      for (int kc = 0; kc < 4; ++kc) bP[kc] = ld_b(sP, 17, kc, lane);
      v8f acc0 = zero8(), acc1 = zero8();
#pragma unroll
      for (int kc = 0; kc < 4; ++kc) {
        acc0 = WMMA4(aC0[kc], bP[kc], acc0);
        acc1 = WMMA4(aC1[kc], bP[kc], acc1);
      }
      st_cd(sCP,            17, lane, acc0);
      st_cd(sCP + 16 * 17,  17, lane, acc1);
    }
    __syncthreads();

    // ---- S = CP @ C^T + R  (32x32, four WMMA tiles, K=16) ----
    {
      v2f aCP0[4], aCP1[4];
#pragma unroll
      for (int kc = 0; kc < 4; ++kc) {
        aCP0[kc] = ld_a(sCP,           17, kc, lane);
        aCP1[kc] = ld_a(sCP + 16 * 17, 17, kc, lane);
      }
      v8f s00 = zero8(), s01 = zero8(), s10 = zero8(), s11 = zero8();
#pragma unroll
      for (int kc = 0; kc < 4; ++kc) {
        s00 = WMMA4(aCP0[kc], bCT0[kc], s00);
        s01 = WMMA4(aCP0[kc], bCT1[kc], s01);
        s10 = WMMA4(aCP1[kc], bCT0[kc], s10);
        s11 = WMMA4(aCP1[kc], bCT1[kc], s11);
      }
      st_cd(sS,               33, lane, s00);
      st_cd(sS + 16,          33, lane, s01);
      st_cd(sS + 16 * 33,     33, lane, s10);
      st_cd(sS + 16 * 33 + 16, 33, lane, s11);
    }
    __syncthreads();
    sS[lane * 33 + lane] += rdiag;                 // S += diag(exp(R_log))
#pragma unroll
    for (int j = 0; j < 16; ++j) sX[lane * 17 + j] = sCP[lane * 17 + j];
    __syncthreads();

    // ---- solve S X = CP for X = K^T (SPD, lane-per-row elimination) ----
    for (int k = 0; k < 32; ++k) {
      float piv = sS[k * 33 + k];
      float f = (lane > k) ? sS[lane * 33 + k] / piv : 0.0f;
      for (int j = k + 1; j < 32; ++j) sS[lane * 33 + j] -= f * sS[k * 33 + j];
#pragma unroll
      for (int j = 0; j < 16; ++j) sX[lane * 17 + j] -= f * sX[k * 17 + j];
      __syncthreads();
    }
    if (lane < 16) {                               // back-sub, lane = RHS column
      for (int k = 31; k >= 0; --k) {
        float s = sX[k * 17 + lane];
        for (int j = k + 1; j < 32; ++j) s -= sS[k * 33 + j] * sX[j * 17 + lane];
        sX[k * 17 + lane] = s / sS[k * 33 + k];
      }
      for (int k = 0; k < 32; ++k) sX[k * 17 + lane] *= m;  // K *= mask
    }
    __syncthreads();

    // ---- mu_t = mu_pred + K r  (lane<16 = x-dim) ----
    if (lane < 16) {
      float acc = sMu[lane];
      for (int b = 0; b < 32; ++b) acc += sX[b * 17 + lane] * sR[b];
      sMuT[lane] = acc;
      o_mu_t[((size_t)t * NUM_SEQ + n) * DIM_X + lane] = acc;
    }
    __syncthreads();

    // ---- Lam_t = P - X^T @ CP  (16x16, K=32 -> 8 WMMA) ----
    v8f ltfrag;
    {
      v8f acc = zero8();
#pragma unroll
      for (int kc = 0; kc < 8; ++kc) {
        v2f ax  = ld_aT(sX,  17, kc, lane);        // (K^T)^T = K rows
        v2f bcp = ld_b (sCP, 17, kc, lane);
        acc = WMMA4(ax, bcp, acc);
      }
      v8f p = ld_cd(sP, 17, lane);
#pragma unroll
      for (int v = 0; v < 8; ++v) ltfrag[v] = p[v] - acc[v];
      st_cd(sT, 17, lane, ltfrag);
      float* o = o_Lt + ((size_t)t * NUM_SEQ + n) * 256;
      st_cd(o, 16, lane, ltfrag);
    }
    __syncthreads();

    // ---- mu_next = A mu_t ----
    if (lane < 16) {
      float acc = 0.0f;
#pragma unroll
      for (int j = 0; j < 16; ++j) acc += Arow[j] * sMuT[j];
      sMu[lane] = acc;                             // becomes next mu_pred
      o_mu_pred[((size_t)t * NUM_SEQ + n) * DIM_X + lane] = acc;
    }

    // ---- T2 = A @ Lam_t ----
    {
      v8f acc = zero8();
#pragma unroll
      for (int kc = 0; kc < 4; ++kc) acc = WMMA4(aA[kc], ld_b(sT, 17, kc, lane), acc);
      st_cd(sT2, 17, lane, acc);
    }
    __syncthreads();

    // ---- Lam_next = T2 @ A^T + W ----
    {
      v8f acc = wfrag;
#pragma unroll
      for (int kc = 0; kc < 4; ++kc) acc = WMMA4(ld_a(sT2, 17, kc, lane), bAT[kc], acc);
      st_cd(sP, 17, lane, acc);                    // becomes next Lambda_pred
      float* o = o_Lp + ((size_t)t * NUM_SEQ + n) * 256;
      st_cd(o, 16, lane, acc);
    }
    __syncthreads();
  }
}

extern "C" void kernel_launch(void* const* d_in, const int* in_sizes, int n_in,
                              void* d_out, int out_size, void* d_ws, size_t ws_size,
                              hipStream_t stream) {
  const float* a    = (const float*)d_in[0];
  const float* mask = (const float*)d_in[1];
  const float* A    = (const float*)d_in[2];
  const float* C    = (const float*)d_in[3];
  const float* mu0  = (const float*)d_in[4];
  const float* Lam0 = (const float*)d_in[5];
  const float* Wlog = (const float*)d_in[6];
  const float* Rlog = (const float*)d_in[7];
  float* out = (float*)d_out;
  ldm_kalman_kernel<<<NUM_SEQ, 32, 0, stream>>>(a, mask, A, C, mu0, Lam0, Wlog, Rlog, out);
}